// GCN_55757265436824
// MI455X (gfx1250) — compile-verified
//
#include <hip/hip_runtime.h>

typedef __attribute__((ext_vector_type(2))) float v2f;
typedef __attribute__((ext_vector_type(8))) float v8f;

#define EPS_BN 1e-5f

// ---------------- elementwise / setup kernels ----------------

__global__ void fill_kernel(float* __restrict__ p, long n, float v) {
  long i = (long)blockIdx.x * blockDim.x + threadIdx.x;
  long s = (long)gridDim.x * blockDim.x;
  for (; i < n; i += s) p[i] = v;
}

__global__ void deg_kernel(const int* __restrict__ dst, float* __restrict__ deg, int E) {
  int i = blockIdx.x * blockDim.x + threadIdx.x;
  if (i < E) atomicAdd(&deg[dst[i]], 1.0f);
}

__global__ void rsqrt_kernel(float* __restrict__ p, int n) {
  int i = blockIdx.x * blockDim.x + threadIdx.x;
  if (i < n) p[i] = rsqrtf(p[i]);
}

__global__ void edgew_kernel(const int* __restrict__ src, const int* __restrict__ dst,
                             const float* __restrict__ dis, float* __restrict__ wgt, int E) {
  int i = blockIdx.x * blockDim.x + threadIdx.x;
  if (i < E) wgt[i] = dis[src[i]] * dis[dst[i]];
}

// ---------------- WMMA fp32 GEMM: out[N x O] = H[N x K] @ Wm[K x O] ----------------
// One wave per 16-row tile. fp32 V_WMMA_F32_16X16X4_F32, K stepped by 4.
// A layout: lane L -> row L%16, v2f = {A[m][k0+2*half], A[m][k0+2*half+1]}, half = L/16.
// B layout mirrors A with N = L%16. C/D: VGPR v holds rows v (lanes 0-15) / v+8 (lanes 16-31).

template <int K, int O>
__global__ void gemm_wmma_kernel(const float* __restrict__ H, const float* __restrict__ Wm,
                                 float* __restrict__ out, int ntiles) {
  int wid = (int)((blockIdx.x * blockDim.x + threadIdx.x) >> 5);
  int lane = threadIdx.x & 31;
  if (wid >= ntiles) return;                  // whole-wave uniform: EXEC stays all-1s
  const int m = lane & 15;
  const int half = lane >> 4;
  const long r0 = (long)wid << 4;

  // Preload the 16xK A panel (K/4 v2f per lane), reused for every 16-col group of W.
  v2f a[K / 4];
#pragma unroll
  for (int kk = 0; kk < K / 4; ++kk) {
    const int ka = kk * 4 + 2 * half;
    const float* ap = &H[(r0 + m) * K + ka];
    a[kk].x = ap[0];
    a[kk].y = ap[1];
  }

#pragma unroll
  for (int n0 = 0; n0 < O; n0 += 16) {
    v8f acc = {};
#pragma unroll
    for (int kk = 0; kk < K / 4; ++kk) {
      const int ka = kk * 4 + 2 * half;
      v2f b;
      b.x = Wm[ka * O + n0 + m];
      b.y = Wm[(ka + 1) * O + n0 + m];
      acc = __builtin_amdgcn_wmma_f32_16x16x4_f32(
          /*neg_a=*/false, a[kk], /*neg_b=*/false, b,
          /*c_mod=*/(short)0, acc, /*reuse_a=*/false, /*reuse_b=*/false);
    }
#pragma unroll
    for (int v = 0; v < 8; ++v) {
      out[(r0 + v + 8 * half) * O + n0 + m] = acc[v];
    }
  }
}

// Scalar tail for rows not covered by 16-row tiles (N=100000 is a multiple of 16,
// but keep it general).
__global__ void gemm_tail_kernel(const float* __restrict__ H, const float* __restrict__ Wm,
                                 float* __restrict__ out, int row0, int rows, int K, int O) {
  int idx = blockIdx.x * blockDim.x + threadIdx.x;
  if (idx >= rows * O) return;
  int r = row0 + idx / O, f = idx % O;
  float acc = 0.f;
  for (int k = 0; k < K; ++k) acc += H[(long)r * K + k] * Wm[k * O + f];
  out[(long)r * O + f] = acc;
}

// ---------------- edge scatter: agg[dst] += hw[src] * wgt ----------------
// Thread = (edge, feature), feature fastest -> coalesced gather/scatter,
// src/dst/wgt broadcast within each O-thread group. hw/agg are L2-resident (25.6MB).

template <int OSH>
__global__ void scatter_kernel(const float* __restrict__ hw, const int* __restrict__ src,
                               const int* __restrict__ dst, const float* __restrict__ wgt,
                               float* __restrict__ agg, int E) {
  const int O = 1 << OSH;
  long idx = (long)blockIdx.x * blockDim.x + threadIdx.x;
  const long stride = (long)gridDim.x * blockDim.x;
  const long total = (long)E << OSH;
  for (; idx < total; idx += stride) {
    const int e = (int)(idx >> OSH);
    const int f = (int)idx & (O - 1);
    if (f == 0) __builtin_prefetch(&src[e + 16384], 0, 0);  // global_prefetch_b8
    const int s = src[e];
    const int d = dst[e];
    const float w = wgt[e];
    atomicAdd(&agg[((long)d << OSH) + f], hw[((long)s << OSH) + f] * w);
  }
}

// ---------------- self-loop + bias + tanh ----------------

template <int OSH>
__global__ void selfloop_tanh_kernel(const float* __restrict__ agg, const float* __restrict__ hw,
                                     const float* __restrict__ dis, const float* __restrict__ bias,
                                     float* __restrict__ hout, long total) {
  const int O = 1 << OSH;
  long idx = (long)blockIdx.x * blockDim.x + threadIdx.x;
  const long stride = (long)gridDim.x * blockDim.x;
  for (; idx < total; idx += stride) {
    const long i = idx >> OSH;
    const int f = (int)idx & (O - 1);
    const float d = dis[i];
    hout[idx] = tanhf(agg[idx] + hw[idx] * d * d + bias[f]);
  }
}

// ---------------- BatchNorm (training-mode, biased variance) ----------------
// Grid stride is a multiple of O so each thread owns exactly one feature column.

template <int OSH>
__global__ void bn_reduce_kernel(const float* __restrict__ h, float* __restrict__ red, long total) {
  const int O = 1 << OSH;
  long i = (long)blockIdx.x * blockDim.x + threadIdx.x;
  const long stride = (long)gridDim.x * blockDim.x;   // multiple of O by launch config
  const int f = (int)(i & (O - 1));
  float sm = 0.f, ss = 0.f;
  for (; i < total; i += stride) {
    const float v = h[i];
    sm += v;
    ss += v * v;
  }
  atomicAdd(&red[f], sm);
  atomicAdd(&red[O + f], ss);
}

template <int OSH>
__global__ void bn_apply_kernel(float* __restrict__ h, const float* __restrict__ red,
                                const float* __restrict__ g, const float* __restrict__ b,
                                long total, float invN) {
  const int O = 1 << OSH;
  long idx = (long)blockIdx.x * blockDim.x + threadIdx.x;
  const long stride = (long)gridDim.x * blockDim.x;
  for (; idx < total; idx += stride) {
    const int f = (int)idx & (O - 1);
    const float m = red[f] * invN;
    const float var = red[O + f] * invN - m * m;
    h[idx] = (h[idx] - m) * rsqrtf(var + EPS_BN) * g[f] + b[f];
  }
}

// ---------------- pooling + readout ----------------

__global__ void pool_kernel(const float* __restrict__ h, const int* __restrict__ bat,
                            float* __restrict__ psum, long total) {
  long idx = (long)blockIdx.x * blockDim.x + threadIdx.x;
  const long stride = (long)gridDim.x * blockDim.x;
  for (; idx < total; idx += stride) {
    const long i = idx >> 6;
    const int f = (int)idx & 63;
    atomicAdd(&psum[((long)bat[i] << 6) + f], h[idx]);
  }
}

__global__ void cnt_kernel(const int* __restrict__ bat, float* __restrict__ pcnt, int n) {
  int i = blockIdx.x * blockDim.x + threadIdx.x;
  if (i < n) atomicAdd(&pcnt[bat[i]], 1.0f);
}

__global__ void readout_kernel(const float* __restrict__ psum, const float* __restrict__ pcnt,
                               const float* __restrict__ Wc, const float* __restrict__ bc,
                               float* __restrict__ out, int G) {
  int g = blockIdx.x * blockDim.x + threadIdx.x;
  if (g >= G) return;
  const float c = fmaxf(pcnt[g], 1.0f);
  float s = 0.f;
  for (int f = 0; f < 64; ++f) s += psum[((long)g << 6) + f] * Wc[f];
  out[g] = s / c + bc[0];
}

// ---------------- host driver ----------------

extern "C" void kernel_launch(void* const* d_in, const int* in_sizes, int n_in,
                              void* d_out, int out_size, void* d_ws, size_t ws_size,
                              hipStream_t stream) {
  (void)n_in; (void)ws_size;
  const float* x   = (const float*)d_in[0];
  const int*   ei  = (const int*)  d_in[1];
  const int*   bat = (const int*)  d_in[2];
  const float* W1  = (const float*)d_in[3];
  const float* b1  = (const float*)d_in[4];
  const float* g1  = (const float*)d_in[5];
  const float* be1 = (const float*)d_in[6];
  const float* W2  = (const float*)d_in[7];
  const float* b2  = (const float*)d_in[8];
  const float* g2  = (const float*)d_in[9];
  const float* be2 = (const float*)d_in[10];
  const float* W3  = (const float*)d_in[11];
  const float* b3  = (const float*)d_in[12];
  const float* Wc  = (const float*)d_in[13];
  const float* bc  = (const float*)d_in[14];
  float* out = (float*)d_out;

  const int N = in_sizes[2];        // batch has N entries
  const int E = in_sizes[1] / 2;    // edge_index is (2, E)
  const int G = out_size;           // pooled output is (G, 1)
  const int* srcI = ei;
  const int* dstI = ei + E;

  // carve workspace
  char* base = (char*)d_ws;
  size_t off = 0;
  auto carve = [&](size_t bytes) -> void* {
    void* p = base + off;
    off += (bytes + 255) & ~(size_t)255;
    return p;
  };
  float* dis  = (float*)carve((size_t)N * 4);            // deg -> rsqrt in place
  float* wgt  = (float*)carve((size_t)E * 4);            // per-edge norm weight
  float* hw   = (float*)carve((size_t)N * 64 * 4);       // h @ W
  float* agg  = (float*)carve((size_t)N * 64 * 4);       // scatter accumulator
  float* h    = (float*)carve((size_t)N * 64 * 4);       // activations
  float* red  = (float*)carve(2 * 64 * 4);               // BN sum / sumsq
  float* psum = (float*)carve((size_t)G * 64 * 4);
  float* pcnt = (float*)carve((size_t)G * 4);

  auto gblk = [](long total) -> unsigned {
    long b = (total + 255) / 256;
    if (b > 65535) b = 65535;
    if (b < 1) b = 1;
    return (unsigned)b;
  };
  auto zero = [&](float* p, long cnt) {
    fill_kernel<<<gblk(cnt), 256, 0, stream>>>(p, cnt, 0.0f);
  };

  // degrees + symmetric normalization weights (shared by all layers)
  fill_kernel<<<gblk(N), 256, 0, stream>>>(dis, (long)N, 1.0f);       // +1 self loop
  deg_kernel<<<(E + 255) / 256, 256, 0, stream>>>(dstI, dis, E);
  rsqrt_kernel<<<(N + 255) / 256, 256, 0, stream>>>(dis, N);
  edgew_kernel<<<(E + 255) / 256, 256, 0, stream>>>(srcI, dstI, dis, wgt, E);

  const int ntiles = N >> 4;
  const int rem = N & 15;
  const unsigned gemm_blocks = (unsigned)((ntiles + 7) / 8);  // 8 waves per 256-thread block

  auto post_gemm = [&](int OSH, const float* bias, const float* gamma, const float* beta) {
    const int O = 1 << OSH;
    const long totalNO = (long)N << OSH;
    zero(agg, totalNO);
    if (OSH == 4)
      scatter_kernel<4><<<4096, 256, 0, stream>>>(hw, srcI, dstI, wgt, agg, E);
    else
      scatter_kernel<6><<<4096, 256, 0, stream>>>(hw, srcI, dstI, wgt, agg, E);
    if (OSH == 4)
      selfloop_tanh_kernel<4><<<gblk(totalNO), 256, 0, stream>>>(agg, hw, dis, bias, h, totalNO);
    else
      selfloop_tanh_kernel<6><<<gblk(totalNO), 256, 0, stream>>>(agg, hw, dis, bias, h, totalNO);
    if (gamma) {
      zero(red, 2 * O);
      const float invN = 1.0f / (float)N;
      if (OSH == 4) {
        bn_reduce_kernel<4><<<512, 256, 0, stream>>>(h, red, totalNO);  // stride % 16 == 0
        bn_apply_kernel<4><<<gblk(totalNO), 256, 0, stream>>>(h, red, gamma, beta, totalNO, invN);
      } else {
        bn_reduce_kernel<6><<<512, 256, 0, stream>>>(h, red, totalNO);  // stride % 64 == 0
        bn_apply_kernel<6><<<gblk(totalNO), 256, 0, stream>>>(h, red, gamma, beta, totalNO, invN);
      }
    }
  };

  // ---- layer 1: 32 -> 16, tanh, BN ----
  gemm_wmma_kernel<32, 16><<<gemm_blocks, 256, 0, stream>>>(x, W1, hw, ntiles);
  if (rem)
    gemm_tail_kernel<<<(rem * 16 + 255) / 256, 256, 0, stream>>>(x, W1, hw, ntiles * 16, rem, 32, 16);
  post_gemm(4, b1, g1, be1);

  // ---- layer 2: 16 -> 64, tanh, BN ----
  gemm_wmma_kernel<16, 64><<<gemm_blocks, 256, 0, stream>>>(h, W2, hw, ntiles);
  if (rem)
    gemm_tail_kernel<<<(rem * 64 + 255) / 256, 256, 0, stream>>>(h, W2, hw, ntiles * 16, rem, 16, 64);
  post_gemm(6, b2, g2, be2);

  // ---- layer 3: 64 -> 64, tanh (no BN) ----
  gemm_wmma_kernel<64, 64><<<gemm_blocks, 256, 0, stream>>>(h, W3, hw, ntiles);
  if (rem)
    gemm_tail_kernel<<<(rem * 64 + 255) / 256, 256, 0, stream>>>(h, W3, hw, ntiles * 16, rem, 64, 64);
  post_gemm(6, b3, nullptr, nullptr);

  // ---- global mean pool + linear readout ----
  zero(psum, (long)G * 64);
  zero(pcnt, (long)G);
  pool_kernel<<<gblk((long)N << 6), 256, 0, stream>>>(h, bat, psum, (long)N << 6);
  cnt_kernel<<<(N + 255) / 256, 256, 0, stream>>>(bat, pcnt, N);
  readout_kernel<<<(G + 255) / 256, 256, 0, stream>>>(psum, pcnt, Wc, bc, out, G);
}